// GPUBiasingMultiModel_28063316313010
// MI455X (gfx1250) — compile-verified
//
#include <hip/hip_runtime.h>
#include <stdint.h>

// n-gram FST advance for MI455X (gfx1250).
// One workgroup per hypothesis. All V labels of a hypothesis share the same
// backoff chain -> per-depth scalar accumulator + priority scatter of arc
// segments replaces 16.7M x 4 global binary searches.
//
// Roofline: output = 2 x B x V x 4B = 128MB streaming stores (~5.5us @
// 23.3TB/s) -> store bound. Arc tables (~7MB) are L2-resident (192MB L2).
// Therefore: maximize waves in flight (LDS trimmed to ~40KB/WG -> 7 WGs/WGP,
// 56 waves), non-temporal output stores so the stream doesn't evict the hot
// arc tables, and async global->LDS (ASYNCcnt path, b128-wide) to stage the
// terminal segment's ilabels while the small higher-order depths execute.

#define THREADS 256
#define VMAX    8192   // VOCAB_SIZE in reference
#define MAXBK   4      // MAX_BACKOFF

struct __align__(16) SmemLayout {
  int           il[VMAX];      // staged ilabels of terminal segment (32KB)
  unsigned char found[VMAX];   // per-label "already matched" flag (8KB)
  int           chain_state[MAXBK];
  float         chain_acc[MAXBK];
  int           seg_lo[MAXBK];
  int           seg_hi[MAXBK];
  int           count;         // labels matched so far
  int           staged_state;  // state whose segment is staged (-1 = none)
  int           staged_lo;
  int           staged_len;
};

__device__ __forceinline__ int lower_bound_i32(const int* __restrict__ a, int n, int key) {
  int lo = 0, hi = n;
  while (lo < hi) {
    int mid = (lo + hi) >> 1;
    if (a[mid] < key) lo = mid + 1; else hi = mid;
  }
  return lo;
}

// CDNA5 async copies, global -> LDS, tracked by ASYNCcnt (not LOADcnt).
// vdst = per-lane LDS byte address, vaddr = 64-bit global address (GV mode).
__device__ __forceinline__ void async_copy_b32(uint32_t lds_byte_addr, const void* gptr) {
  uint64_t ga = (uint64_t)(uintptr_t)gptr;
  asm volatile("global_load_async_to_lds_b32 %0, %1, off"
               :: "v"(lds_byte_addr), "v"(ga) : "memory");
}
__device__ __forceinline__ void async_copy_b128(uint32_t lds_byte_addr, const void* gptr) {
  uint64_t ga = (uint64_t)(uintptr_t)gptr;
  asm volatile("global_load_async_to_lds_b128 %0, %1, off"
               :: "v"(lds_byte_addr), "v"(ga) : "memory");
}
__device__ __forceinline__ void wait_asynccnt0() {
  asm volatile("s_wait_asynccnt 0" ::: "memory");
}

__global__ void __launch_bounds__(THREADS)
ngram_advance_kernel(const float* __restrict__ arc_w,   // [A]
                     const int*   __restrict__ arc_to,  // [A]
                     const int*   __restrict__ arc_from,// [A] sorted
                     const int*   __restrict__ arc_il,  // [A] sorted within state
                     const int*   __restrict__ bk_to,   // [NS]
                     const float* __restrict__ bk_w,    // [NS]
                     const float* __restrict__ fin_w,   // [NS]
                     const float* __restrict__ alpha,   // [M]
                     const int*   __restrict__ states,  // [B]
                     const int*   __restrict__ mids,    // [B]
                     const int*   __restrict__ eos_p,   // [1]
                     float* __restrict__ scores_out,    // [B*V]
                     float* __restrict__ next_out,      // [B*V] (float-encoded)
                     int A, int V)
{
  __shared__ SmemLayout sm;
  const int   b   = blockIdx.x;
  const int   tid = threadIdx.x;
  const int   st0 = states[b];
  const float al  = alpha[mids[b]];
  float* __restrict__ srow = scores_out + (size_t)b * (size_t)V;
  float* __restrict__ nrow = next_out   + (size_t)b * (size_t)V;

  // ---- init found[] (packed u32 writes) + walk the backoff chain ----
  {
    uint32_t* f32p = (uint32_t*)sm.found;
    for (int j = tid; j < VMAX / 4; j += THREADS) f32p[j] = 0u;
  }
  if (tid == 0) {
    sm.count = 0;
    int cur = st0;
    float acc = 0.0f;
    int stg = -1;
    for (int d = 0; d < MAXBK; ++d) {
      sm.chain_state[d] = cur;
      sm.chain_acc[d]   = acc;                    // accum BEFORE this depth's lookup
      int lo = lower_bound_i32(arc_from, A, cur);
      int hi = lower_bound_i32(arc_from, A, cur + 1);
      sm.seg_lo[d] = lo;
      sm.seg_hi[d] = hi;
      if (stg < 0) {
        int len = hi - lo;
        if (len >= 1024 && len <= VMAX) { stg = cur; sm.staged_lo = lo; sm.staged_len = len; }
      }
      acc += bk_w[cur];                           // applied only on miss = later depths
      cur  = bk_to[cur];
    }
    sm.staged_state = stg;
  }
  __syncthreads();

  // ---- kick off async staging of terminal-segment ilabels into LDS ----
  const int stg_state = sm.staged_state;
  if (stg_state >= 0) {
    const int lo  = sm.staged_lo;
    const int len = sm.staged_len;
    if ((lo & 3) == 0) {
      // 16B-aligned fast path: one b128 per lane per chunk (512B/wave-instr)
      const int n4 = len >> 2;
      for (int c = tid; c < n4; c += THREADS)
        async_copy_b128((uint32_t)(uintptr_t)&sm.il[c * 4], &arc_il[lo + c * 4]);
      for (int i = (n4 << 2) + tid; i < len; i += THREADS)
        async_copy_b32((uint32_t)(uintptr_t)&sm.il[i], &arc_il[lo + i]);
    } else {
      for (int i = tid; i < len; i += THREADS)
        async_copy_b32((uint32_t)(uintptr_t)&sm.il[i], &arc_il[lo + i]);
    }
  }

  // ---- depth loop: priority scatter (earliest depth wins; leftmost arc wins) ----
  bool lds_ready = false;
  for (int d = 0; d < MAXBK; ++d) {
    __syncthreads();                 // prior depth's found[]/count adds complete
    const int cnt = sm.count;
    __syncthreads();                 // everyone read cnt before any new writes
    if (cnt >= V) break;             // uniform: all labels matched

    const int   cs   = sm.chain_state[d];
    const int   lo   = sm.seg_lo[d];
    const int   len  = sm.seg_hi[d] - lo;
    const float acc  = sm.chain_acc[d];
    const bool  use_lds = (stg_state >= 0) && (cs == stg_state);  // block-uniform

    if (use_lds && !lds_ready) {
      wait_asynccnt0();              // each wave waits for its own copies
      __syncthreads();               // ...then all waves' data is in LDS
      lds_ready = true;
    }

    int newfound = 0;
    for (int i = tid; i < len; i += THREADS) {
      int il, prev_il = -1;
      if (use_lds) {
        il = sm.il[i];
        if (i > 0) prev_il = sm.il[i - 1];
      } else {
        il = arc_il[lo + i];
        if (i > 0) prev_il = arc_il[lo + i - 1];
      }
      // ilabels sorted in segment -> duplicates contiguous -> leftmost-of-run
      // filter makes winners unique within this depth (no LDS races).
      if ((unsigned)il < (unsigned)VMAX && (i == 0 || prev_il != il) && !sm.found[il]) {
        sm.found[il] = 1;
        ++newfound;
        const float w  = arc_w[lo + i];   // L2-hot, read once
        const int   to = arc_to[lo + i];  // L2-hot, read once
        __builtin_nontemporal_store((acc + w) * al, &srow[il]);
        __builtin_nontemporal_store((float)to,      &nrow[il]);
      }
    }
    if (newfound) atomicAdd(&sm.count, newfound);
  }
  __syncthreads();

  // ---- zero-fill never-matched labels (packed u32 scan of found[]) ----
  {
    const uint32_t* f32p = (const uint32_t*)sm.found;
    for (int j = tid; j < VMAX / 4; j += THREADS) {
      const uint32_t f4 = f32p[j];
      if (f4 != 0x01010101u) {
        const int vbase = j * 4;
        #pragma unroll
        for (int k = 0; k < 4; ++k) {
          if (((f4 >> (8 * k)) & 0xFFu) == 0u && vbase + k < V) {
            __builtin_nontemporal_store(0.0f, &srow[vbase + k]);
            __builtin_nontemporal_store(0.0f, &nrow[vbase + k]);
          }
        }
      }
    }
  }
  __syncthreads();

  // ---- eos override: final weight, stay in current state ----
  if (tid == 0) {
    const int eos = *eos_p;
    if ((unsigned)eos < (unsigned)V) {
      __builtin_nontemporal_store(fin_w[st0] * al, &srow[eos]);
      __builtin_nontemporal_store((float)st0,      &nrow[eos]);
    }
  }
}

extern "C" void kernel_launch(void* const* d_in, const int* in_sizes, int n_in,
                              void* d_out, int out_size, void* d_ws, size_t ws_size,
                              hipStream_t stream) {
  (void)n_in; (void)d_ws; (void)ws_size; (void)out_size;
  const float* arc_w    = (const float*)d_in[0];
  const int*   arc_to   = (const int*)  d_in[1];
  const int*   arc_from = (const int*)  d_in[2];
  const int*   arc_il   = (const int*)  d_in[3];
  const int*   bk_to    = (const int*)  d_in[4];
  const float* bk_w     = (const float*)d_in[5];
  const float* fin_w    = (const float*)d_in[6];
  const float* alpha    = (const float*)d_in[7];
  const int*   states   = (const int*)  d_in[8];
  const int*   mids     = (const int*)  d_in[9];
  const int*   eos_p    = (const int*)  d_in[10];

  const int A = in_sizes[0];   // number of arcs
  const int B = in_sizes[8];   // batch (hypotheses)
  const int V = VMAX;          // VOCAB_SIZE (fixed in reference)

  float* scores = (float*)d_out;                    // [B*V] f32
  float* nexts  = scores + (size_t)B * (size_t)V;   // [B*V] float-encoded ints

  ngram_advance_kernel<<<dim3(B), dim3(THREADS), 0, stream>>>(
      arc_w, arc_to, arc_from, arc_il, bk_to, bk_w, fin_w, alpha,
      states, mids, eos_p, scores, nexts, A, V);
}